// NMS_76115410420242
// MI455X (gfx1250) — compile-verified
//
#include <hip/hip_runtime.h>
#include <stdint.h>

#define BATCH   8
#define NPTS    65536
#define NCLS    80
#define NROWS   (BATCH * NPTS)      // 524288
#define TOPK    4096
#define CAP     16384               // sort capacity; expected valid ~1300
#define CONF_T  4.0f
#define NMS_T   0.5f
#define SPAD    84                  // LDS row pitch in floats (16B aligned, 2-way bank conflict)

typedef __attribute__((ext_vector_type(2))) float v2f;
typedef __attribute__((ext_vector_type(8))) float v8f;

// ---------------- workspace layout (bytes) ----------------
static constexpr size_t OFF_COUNT = 0;
static constexpr size_t OFF_VKEY  = 16;                            // u64[CAP] 128KB
static constexpr size_t OFF_TBOX  = OFF_VKEY  + (size_t)CAP * 8;   // float4[TOPK]
static constexpr size_t OFF_TAREA = OFF_TBOX  + (size_t)TOPK * 16; // float[TOPK]
static constexpr size_t OFF_TVALS = OFF_TAREA + (size_t)TOPK * 4;
static constexpr size_t OFF_TCT   = OFF_TVALS + (size_t)TOPK * 4;  // int[TOPK]
static constexpr size_t OFF_TCL   = OFF_TCT   + (size_t)TOPK * 4;
static constexpr size_t OFF_TIM   = OFF_TCL   + (size_t)TOPK * 4;
static constexpr size_t OFF_TVK   = OFF_TIM   + (size_t)TOPK * 4;
static constexpr size_t OFF_SUP   = OFF_TVK   + (size_t)TOPK * 4;  // u16[TOPK*256] = 2MB

__global__ void nms_init(int* count) { *count = 0; }

// ------- pass 1: stage logits via async global->LDS, max/argmax, filter, pack key -------
// key = score_bits<<32 | (0xFFFFF - idx)<<12 | cls  (desc sort => score desc, idx asc)
__global__ void nms_score(const float* __restrict__ logits,
                          unsigned long long* __restrict__ vkey,
                          int* __restrict__ count) {
  // 8 waves * 32 rows * 84 floats = 86016 B static LDS
  __shared__ float smem[8 * 32 * SPAD];
  const int wave = threadIdx.x >> 5;
  const int lane = threadIdx.x & 31;
  const int rowBase = blockIdx.x * blockDim.x + wave * 32;   // wave owns 32 rows
  float* wbuf = smem + wave * 32 * SPAD;
  const char* gbase = (const char*)(logits + (size_t)rowBase * NCLS);

  // Stage 32 rows (32*20 float4 = 10KB) with lane-contiguous 512B requests.
  // ASYNCcnt path: global_load_async_to_lds_b128 (per-lane LDS scatter w/ pitch pad).
#pragma unroll
  for (int it = 0; it < 20; ++it) {
    int q = it * 32 + lane;                 // float4 index within the wave's chunk
    int r = q / 20, c = q % 20;             // target row / float4-column
    unsigned ldsb = (unsigned)(uintptr_t)(&wbuf[r * SPAD + c * 4]); // LDS byte offset
    unsigned goff = (unsigned)(q * 16);     // byte offset from gbase
    asm volatile("global_load_async_to_lds_b128 %0, %1, %2"
                 :: "v"(ldsb), "v"(goff), "s"(gbase) : "memory");
  }
  asm volatile("s_wait_asynccnt 0x0" ::: "memory");

  // Each lane reduces its own row out of LDS (first-occurrence argmax).
  const float4* rp = (const float4*)(wbuf + lane * SPAD);   // 336B pitch, 16B aligned
  float best = -3.402823466e+38f;
  int bi = 0;
#pragma unroll 4
  for (int k = 0; k < NCLS / 4; ++k) {
    float4 v = rp[k];
    if (v.x > best) { best = v.x; bi = 4 * k;     }
    if (v.y > best) { best = v.y; bi = 4 * k + 1; }
    if (v.z > best) { best = v.z; bi = 4 * k + 2; }
    if (v.w > best) { best = v.w; bi = 4 * k + 3; }
  }
  int row = rowBase + lane;
  if (best > CONF_T) {
    int pos = atomicAdd(count, 1);
    if (pos < CAP) {
      unsigned fb  = __float_as_uint(best);
      unsigned inv = 0xFFFFFu - (unsigned)row;
      vkey[pos] = ((unsigned long long)fb << 32) |
                  ((unsigned long long)inv << 12) | (unsigned)bi;
    }
  }
}

// ------- pass 2: single-WG bitonic sort of CAP u64 keys in LDS, gather topk -------
__global__ void nms_sort_gather(const int* __restrict__ count,
                                const unsigned long long* __restrict__ vkey,
                                const float* __restrict__ boxes,
                                char* __restrict__ ws) {
  extern __shared__ unsigned long long s_key[];   // CAP * 8 = 128KB (<320KB/WGP)
  const int n = CAP;
  int cnt = *count; if (cnt > CAP) cnt = CAP;
  for (int i = threadIdx.x; i < n; i += blockDim.x)
    s_key[i] = (i < cnt) ? vkey[i] : 0ULL;        // 0 sorts last (valid keys >0)
  __syncthreads();
  for (int k = 2; k <= n; k <<= 1) {
    for (int j = k >> 1; j > 0; j >>= 1) {
      for (int i = threadIdx.x; i < n; i += blockDim.x) {
        int p = i ^ j;
        if (p > i) {
          unsigned long long a = s_key[i], b = s_key[p];
          bool desc = ((i & k) == 0);
          if ((b > a) == desc) { s_key[i] = b; s_key[p] = a; }
        }
      }
      __syncthreads();
    }
  }
  float4* tbox  = (float4*)(ws + OFF_TBOX);
  float*  tarea = (float*)(ws + OFF_TAREA);
  float*  tvals = (float*)(ws + OFF_TVALS);
  int*    tct   = (int*)(ws + OFF_TCT);
  int*    tcl   = (int*)(ws + OFF_TCL);
  int*    tim   = (int*)(ws + OFF_TIM);
  int*    tvk   = (int*)(ws + OFF_TVK);
  for (int t = threadIdx.x; t < TOPK; t += blockDim.x) {
    unsigned long long key = s_key[t];
    bool valid = (key != 0ULL);
    int cls = (int)(key & 0xFFF);
    int idx = 0xFFFFF - (int)((key >> 12) & 0xFFFFF);
    float sc = __uint_as_float((unsigned)(key >> 32));
    float4 bx = make_float4(0.f, 0.f, 0.f, 0.f);
    int img = -1, cat = -1; float area = 0.f;
    if (valid) {
      bx   = ((const float4*)boxes)[idx];
      img  = idx >> 16;                 // idx / NPTS
      cat  = cls + img * NCLS;
      area = (bx.z - bx.x) * (bx.w - bx.y);
    }
    tbox[t] = bx; tarea[t] = area; tvals[t] = valid ? sc : -1e9f;
    tct[t] = cat; tcl[t] = valid ? cls : -1; tim[t] = img; tvk[t] = valid ? 1 : 0;
  }
}

// ------- pass 3: 16x16-tile suppression matrix; WMMA computes area_i+area_j -------
__global__ void nms_supmat(const float4* __restrict__ tbox,
                           const float*  __restrict__ tarea,
                           const int*    __restrict__ tct,
                           unsigned short* __restrict__ sup) {
  int wave = blockIdx.x * (blockDim.x >> 5) + (threadIdx.x >> 5); // 0..65535
  int lane = threadIdx.x & 31;
  int ti = wave >> 8, tj = wave & 255;
  int i0 = ti << 4,  j0 = tj << 4;
  bool lo = (lane < 16);
  int nidx = lane & 15;
  // A(16x4) row m: [area_i, 1, 0, 0]; B(4x16) col n: [1, area_j, 0, 0]^T
  v2f a, b;
  a.x = lo ? tarea[i0 + nidx] : 0.f;   a.y = lo ? 1.0f : 0.f;
  b.x = lo ? 1.0f : 0.f;               b.y = lo ? tarea[j0 + nidx] : 0.f;
  v8f acc = {};
  acc = __builtin_amdgcn_wmma_f32_16x16x4_f32(false, a, false, b,
                                              (short)0, acc, false, false);
  // acc[v] = area_i[m] + area_j[n],  m = v + (lane>=16 ? 8 : 0), n = lane&15
  int j = j0 + nidx;
  float4 bj = tbox[j];
  int ctj = tct[j];
  int mbase = lo ? 0 : 8;
#pragma unroll
  for (int v = 0; v < 8; ++v) {
    int irow = i0 + mbase + v;
    float4 bi = tbox[irow];
    float iw = fmaxf(fminf(bi.z, bj.z) - fmaxf(bi.x, bj.x), 0.f);
    float ih = fmaxf(fminf(bi.w, bj.w) - fmaxf(bi.y, bj.y), 0.f);
    float inter = iw * ih;
    float uni   = acc[v] - inter;
    float iou   = inter / fmaxf(uni, 1e-12f);
    bool pred = (tct[irow] == ctj) && (iou > NMS_T);
    unsigned mask = __builtin_amdgcn_ballot_w32(pred);
    if (lane == 0) {
      sup[(size_t)(i0 + v)     * 256 + tj] = (unsigned short)(mask & 0xFFFFu);
      sup[(size_t)(i0 + v + 8) * 256 + tj] = (unsigned short)(mask >> 16);
    }
  }
}

// ------- pass 4: sequential greedy NMS over bitmask + output writeback -------
__global__ void nms_sweep(const unsigned short* __restrict__ sup,
                          const int* __restrict__ tvk,
                          const float4* __restrict__ tbox,
                          const float* __restrict__ tvals,
                          const int* __restrict__ tcl,
                          const int* __restrict__ tim,
                          float* __restrict__ out) {
  __shared__ unsigned s_keep[TOPK / 32];  // 128 words
  int t = threadIdx.x;                    // 128 threads
  unsigned w = 0;
  for (int bb = 0; bb < 32; ++bb) w |= (tvk[t * 32 + bb] ? 1u : 0u) << bb;
  s_keep[t] = w;
  for (int i = 0; i < TOPK; ++i) {
    __syncthreads();
    bool ki = (s_keep[i >> 5] >> (i & 31)) & 1u;   // uniform
    if (ki) {
      const unsigned* row = (const unsigned*)(sup + (size_t)i * 256);
      unsigned m = row[t];
      int wi = i >> 5, bi = i & 31;
      if (t < wi) m = 0;
      else if (t == wi) {
        unsigned low = (bi == 31) ? 0xFFFFFFFFu : ((2u << bi) - 1u);
        m &= ~low;                                  // only order > i suppresses
      }
      s_keep[t] &= ~m;
    }
  }
  __syncthreads();
  float* out_img = out;
  float* out_box = out + TOPK;
  float* out_cls = out + TOPK + TOPK * 4;
  float* out_sco = out + TOPK + TOPK * 4 + TOPK;
  for (int q = t; q < TOPK; q += blockDim.x) {
    bool k = (s_keep[q >> 5] >> (q & 31)) & 1u;
    float4 bx = tbox[q];
    out_img[q]         = k ? (float)tim[q] : -1.0f;
    out_box[q * 4 + 0] = k ? bx.x : 0.f;
    out_box[q * 4 + 1] = k ? bx.y : 0.f;
    out_box[q * 4 + 2] = k ? bx.z : 0.f;
    out_box[q * 4 + 3] = k ? bx.w : 0.f;
    out_cls[q]         = k ? (float)tcl[q] : -1.0f;
    out_sco[q]         = k ? tvals[q] : 0.0f;
  }
}

extern "C" void kernel_launch(void* const* d_in, const int* in_sizes, int n_in,
                              void* d_out, int out_size, void* d_ws, size_t ws_size,
                              hipStream_t stream) {
  (void)in_sizes; (void)n_in; (void)out_size; (void)ws_size;
  const float* logits = (const float*)d_in[0];
  const float* boxes  = (const float*)d_in[1];
  char* ws = (char*)d_ws;
  int* count = (int*)(ws + OFF_COUNT);
  unsigned long long* vkey = (unsigned long long*)(ws + OFF_VKEY);
  float4* tbox  = (float4*)(ws + OFF_TBOX);
  float*  tarea = (float*)(ws + OFF_TAREA);
  int*    tct   = (int*)(ws + OFF_TCT);
  float*  tvals = (float*)(ws + OFF_TVALS);
  int*    tcl   = (int*)(ws + OFF_TCL);
  int*    tim   = (int*)(ws + OFF_TIM);
  int*    tvk   = (int*)(ws + OFF_TVK);
  unsigned short* sup = (unsigned short*)(ws + OFF_SUP);

  nms_init<<<1, 1, 0, stream>>>(count);
  nms_score<<<NROWS / 256, 256, 0, stream>>>(logits, vkey, count);
  nms_sort_gather<<<1, 1024, (size_t)CAP * 8, stream>>>(count, vkey, boxes, ws);
  // 256x256 tiles, 1 wave each, 8 waves/block -> 8192 blocks
  nms_supmat<<<(256 * 256) / 8, 256, 0, stream>>>(tbox, tarea, tct, sup);
  nms_sweep<<<1, 128, 0, stream>>>(sup, tvk, tbox, tvals, tcl, tim, (float*)d_out);
}